// PoseHead_50405736186595
// MI455X (gfx1250) — compile-verified
//
#include <hip/hip_runtime.h>
#include <hip/hip_bf16.h>
#include <math.h>

// ---------------------------------------------------------------------------
// PoseHead: camera translation from flow + depth + IMU.
// Reduces to per-batch 8-scalar reduction (AtA[4 unique], Atb[3], btb),
// then a 3x3 Cramer solve + residual norm.
// Kernel 1: bandwidth-bound strided gather + per-lane f32 accumulation.
// Kernel 2: partial-sum reduction expressed as onesT x P via
//           V_WMMA_F32_16X16X4_F32 (f32 WMMA = reference precision), then solve.
// ---------------------------------------------------------------------------

typedef float v2f __attribute__((ext_vector_type(2)));
typedef float v8f __attribute__((ext_vector_type(8)));

#define CUTK   20
#define IMH    480
#define IMW    640
#define HWSZ   (IMH * IMW)
#define JW     60            // sampled cols per row (600 / 10)
#define NROWS  440           // 480 - 2*20
#define NSAMP  (NROWS * JW)  // 26400 samples per batch
#define CHUNKS 32            // blocks per batch in stage 1
#define BLOCK  256           // 8 wave32s

// Stage 1: per-(batch, chunk) partial sums of the 8 independent accumulators.
// Partial layout: partials[b][chunk][16] (cols 8..15 zero-padded for WMMA B).
__global__ __launch_bounds__(BLOCK) void pose_partial_kernel(
    const float* __restrict__ flow,   // (B, 2, H, W)
    const float* __restrict__ depth,  // (B, H, W)
    const float* __restrict__ fl,     // (B,)
    const float* __restrict__ gyro,   // (B, 3)
    float* __restrict__ partials)     // (B, CHUNKS, 16)
{
    const int b     = blockIdx.y;
    const int chunk = blockIdx.x;
    const int tid   = threadIdx.x;

    const float F    = fl[b];
    const float invF = 1.0f / F;
    const float w0 = gyro[3 * b + 0];
    const float w1 = gyro[3 * b + 1];
    const float w2 = gyro[3 * b + 2];

    const float* __restrict__ dp  = depth + (size_t)b * HWSZ;
    const float* __restrict__ fxp = flow + (size_t)b * 2 * HWSZ;
    const float* __restrict__ fyp = fxp + HWSZ;

    // Accumulators: S00 (= AtA00 = AtA11), S02, S12, S22, T0, T1, T2, BB.
    float S00 = 0.f, S02 = 0.f, S12 = 0.f, S22 = 0.f;
    float T0 = 0.f, T1 = 0.f, T2 = 0.f, BB = 0.f;

    const int stride = CHUNKS * BLOCK;
    for (int s = chunk * BLOCK + tid; s < NSAMP; s += stride) {
        const int ii  = s / JW;          // 0..439
        const int jj  = s - ii * JW;     // 0..59
        const int off = (CUTK + ii) * IMW + (CUTK + 10 * jj);

        // Prefetch next tranche (stride-40B gather: warm the sectors early).
        const int sn = s + stride;
        if (sn < NSAMP) {
            const int ii2  = sn / JW;
            const int jj2  = sn - ii2 * JW;
            const int off2 = (CUTK + ii2) * IMW + (CUTK + 10 * jj2);
            __builtin_prefetch(dp + off2, 0, 1);
            __builtin_prefetch(fxp + off2, 0, 1);
            __builtin_prefetch(fyp + off2, 0, 1);
        }

        const float d  = dp[off];
        const float fx = fxp[off];
        const float fy = fyp[off];

        const float u = (float)(10 * jj - 300);  // j - W/2
        const float v = (float)(ii - 220);       // i - H/2

        // b_corr = flow - A_rot @ omega
        const float uvF = u * v * invF;
        const float bx  = fx - (uvF * w0 - (F + u * u * invF) * w1 + v * w2);
        const float by  = fy - ((F + v * v * invF) * w0 - uvF * w1 - u * w2);

        const float id = 1.0f / d;
        const float sF = F * id;   // fl / depth
        const float pu = u * id;
        const float pv = v * id;

        // x-row of A_tr: (-sF, 0, pu);  y-row: (0, -sF, pv)
        S00 = fmaf(sF, sF, S00);                       // shared by AtA00 & AtA11
        S02 = fmaf(-sF, pu, S02);
        S12 = fmaf(-sF, pv, S12);
        S22 = fmaf(pu, pu, fmaf(pv, pv, S22));
        T0  = fmaf(-sF, bx, T0);
        T1  = fmaf(-sF, by, T1);
        T2  = fmaf(pu, bx, fmaf(pv, by, T2));
        BB  = fmaf(bx, bx, fmaf(by, by, BB));
    }

    float acc[8] = {S00, S02, S12, S22, T0, T1, T2, BB};
#pragma unroll
    for (int i = 0; i < 8; ++i) {
#pragma unroll
        for (int m = 16; m >= 1; m >>= 1)
            acc[i] += __shfl_xor(acc[i], m, 32);
    }

    __shared__ float lsum[8][8];  // [wave][accumulator]
    const int wave = tid >> 5;
    const int lane = tid & 31;
    if (lane == 0) {
#pragma unroll
        for (int i = 0; i < 8; ++i) lsum[wave][i] = acc[i];
    }
    __syncthreads();

    if (tid < 16) {
        float v = 0.f;
        if (tid < 8) {
#pragma unroll
            for (int w = 0; w < 8; ++w) v += lsum[w][tid];
        }
        partials[((size_t)b * CHUNKS + chunk) * 16 + tid] = v;  // cols 8..15 = 0
    }
}

// Stage 2: one wave32 per batch. total = onesT x P via V_WMMA_F32_16X16X4_F32
// (A = 16x4 ones, B = 4 partial rows per step, C accumulates over CHUNKS/4
// steps; D row M=0 lane n = sum of column n). Then Cramer solve + residual.
__global__ __launch_bounds__(32) void pose_solve_kernel(
    const float* __restrict__ partials,  // (B, CHUNKS, 16)
    float* __restrict__ out,             // pose (B,3) then res (B,)
    int B)
{
    const int b    = blockIdx.x;
    const int lane = threadIdx.x;

    const float* __restrict__ P = partials + (size_t)b * CHUNKS * 16;

    const v2f onesA = {1.0f, 1.0f};  // A[m][k] = 1 for all lanes/VGPRs
    v8f c = {};

    const int col = lane & 15;                 // B-matrix column N
    const int k0  = (lane < 16) ? 0 : 2;       // K rows held by this lane half
#pragma unroll
    for (int r = 0; r < CHUNKS / 4; ++r) {
        const int row = 4 * r + k0;
        v2f bB;
        bB.x = P[row * 16 + col];        // B[k0+0][col]
        bB.y = P[(row + 1) * 16 + col];  // B[k0+1][col]
        // D = A*B + C  (f32 in/out, K=4 rows of partials per issue)
        c = __builtin_amdgcn_wmma_f32_16x16x4_f32(
                /*neg_a=*/false, onesA, /*neg_b=*/false, bB,
                /*c_mod=*/(short)0, c, /*reuse_a=*/false, /*reuse_b=*/false);
    }

    // D VGPR0, lanes 0..15 = (M=0, N=lane): column totals.
    const float colsum = c[0];
    const float a   = __shfl(colsum, 0, 32);  // S00 (= AtA00 = AtA11)
    const float cc  = __shfl(colsum, 1, 32);  // S02
    const float dd  = __shfl(colsum, 2, 32);  // S12
    const float e   = __shfl(colsum, 3, 32);  // S22
    const float Tb0 = __shfl(colsum, 4, 32);
    const float Tb1 = __shfl(colsum, 5, 32);
    const float Tb2 = __shfl(colsum, 6, 32);
    const float BB  = __shfl(colsum, 7, 32);

    if (lane == 0) {
        // AtA = [[a,0,cc],[0,a,dd],[cc,dd,e]] — symmetric, Cramer via adjugate.
        const float adj00 = a * e - dd * dd;
        const float adj01 = cc * dd;
        const float adj02 = -a * cc;
        const float adj11 = a * e - cc * cc;
        const float adj12 = -a * dd;
        const float adj22 = a * a;
        const float det   = a * (a * e - dd * dd - cc * cc);
        const float inv   = 1.0f / det;

        const float t0 = (adj00 * Tb0 + adj01 * Tb1 + adj02 * Tb2) * inv;
        const float t1 = (adj01 * Tb0 + adj11 * Tb1 + adj12 * Tb2) * inv;
        const float t2 = (adj02 * Tb0 + adj12 * Tb1 + adj22 * Tb2) * inv;

        // ||A_tr t - b_corr||^2 = tT AtA t - 2 tT Atb + bTb
        float q = a * (t0 * t0 + t1 * t1) + e * t2 * t2
                + 2.0f * (cc * t0 * t2 + dd * t1 * t2)
                - 2.0f * (t0 * Tb0 + t1 * Tb1 + t2 * Tb2) + BB;
        q = q > 0.f ? q : 0.f;

        out[b * 3 + 0] = t0;
        out[b * 3 + 1] = t1;
        out[b * 3 + 2] = t2;
        out[3 * B + b] = sqrtf(q);
    }
}

extern "C" void kernel_launch(void* const* d_in, const int* in_sizes, int n_in,
                              void* d_out, int out_size, void* d_ws, size_t ws_size,
                              hipStream_t stream) {
    const float* flow  = (const float*)d_in[0];  // (B,2,480,640)
    const float* depth = (const float*)d_in[1];  // (B,480,640)
    const float* fl    = (const float*)d_in[2];  // (B,)
    const float* gyro  = (const float*)d_in[3];  // (B,3)
    float* out = (float*)d_out;

    const int B = in_sizes[2];  // fl has B elements
    float* partials = (float*)d_ws;  // B * CHUNKS * 16 floats (64 KB @ B=32)

    dim3 grid1(CHUNKS, B);
    pose_partial_kernel<<<grid1, BLOCK, 0, stream>>>(flow, depth, fl, gyro, partials);
    pose_solve_kernel<<<B, 32, 0, stream>>>(partials, out, B);
}